// MySGCL_36361193128546
// MI455X (gfx1250) — compile-verified
//
#include <hip/hip_runtime.h>
#include <stdint.h>

typedef __attribute__((ext_vector_type(2)))  float    v2f;
typedef __attribute__((ext_vector_type(8)))  float    v8f;

#define IN_F  128
#define OUT_F 64

// Optional gfx1250 async global->LDS path (guarded; falls back to float4 copy)
#if defined(__has_builtin)
#if __has_builtin(__builtin_amdgcn_global_load_async_to_lds_b128) && \
    __has_builtin(__builtin_amdgcn_s_wait_asynccnt)
#define HAVE_ASYNC_LDS 1
#endif
#endif

#if defined(HAVE_ASYNC_LDS)
// Builtin signature (from hipcc diagnostic): param0 = v4i in AS1 (global),
// param1 = v4i in AS3 (LDS), then two constant ints (offset, cpol).
typedef int v4i_b __attribute__((vector_size(16)));
typedef __attribute__((address_space(1))) v4i_b as1_v4i;
typedef __attribute__((address_space(3))) v4i_b as3_v4i;
#endif

// ---------------------------------------------------------------------------
// integer mixer (splitmix-style finalizer) for deterministic augmentation
// ---------------------------------------------------------------------------
__device__ __forceinline__ uint32_t mix32(uint32_t x) {
  x ^= x >> 16; x *= 0x7feb352dU;
  x ^= x >> 15; x *= 0x846ca68bU;
  x ^= x >> 16;
  return x;
}

// ---------------------------------------------------------------------------
// Build the 4 augmented edge sets:
//   v=0 con_pos : pos edges, 10% resampled uniformly
//   v=1 sig_pos : pos edges, 10% swapped with neg
//   v=2 con_neg : neg edges, 10% resampled uniformly
//   v=3 sig_neg : neg edges, 10% swapped with pos
// Layout: ev[v*2*M + e] = src, ev[v*2*M + M + e] = dst
// ---------------------------------------------------------------------------
__global__ void build_edges_kernel(const int* __restrict__ pos,
                                   const int* __restrict__ neg,
                                   int* __restrict__ ev, int M, int N) {
  const uint32_t SEED = 42u * 0x9e3779b9u;
  long long stride = (long long)gridDim.x * blockDim.x;
  for (long long e = (long long)blockIdx.x * blockDim.x + threadIdx.x; e < M;
       e += stride) {
    int ps = pos[e], pd = pos[M + e];
    int ns = neg[e], nd = neg[M + e];
    uint32_t r0 = mix32((uint32_t)e * 4u + 0u + SEED);
    uint32_t r1 = mix32((uint32_t)e * 4u + 1u + SEED);
    uint32_t r2 = mix32((uint32_t)e * 4u + 2u + SEED);
    uint32_t r3 = mix32((uint32_t)e * 4u + 3u + SEED);
    // con_pos
    int s = ps, d = pd;
    if (r0 % 10u == 0u) {
      s = (int)(mix32(r0 ^ 0xA511E9B3u) % (uint32_t)N);
      d = (int)(mix32(r0 ^ 0x63D83595u) % (uint32_t)N);
    }
    ev[0 * 2 * M + e] = s;  ev[0 * 2 * M + M + e] = d;
    // sig_pos
    bool sw1 = (r1 % 10u == 0u);
    ev[1 * 2 * M + e]     = sw1 ? ns : ps;
    ev[1 * 2 * M + M + e] = sw1 ? nd : pd;
    // con_neg
    s = ns; d = nd;
    if (r2 % 10u == 0u) {
      s = (int)(mix32(r2 ^ 0x2545F491u) % (uint32_t)N);
      d = (int)(mix32(r2 ^ 0x8F1BBCDCu) % (uint32_t)N);
    }
    ev[2 * 2 * M + e] = s;  ev[2 * 2 * M + M + e] = d;
    // sig_neg
    bool sw3 = (r3 % 10u == 0u);
    ev[3 * 2 * M + e]     = sw3 ? ps : ns;
    ev[3 * 2 * M + M + e] = sw3 ? pd : nd;
  }
}

// ---------------------------------------------------------------------------
__global__ void fill_kernel(float* __restrict__ p, float v, long long n) {
  long long stride = (long long)gridDim.x * blockDim.x;
  for (long long i = (long long)blockIdx.x * blockDim.x + threadIdx.x; i < n;
       i += stride)
    p[i] = v;
}

// deg[v][i] = 1 (self loop) + #incoming edges;  blockIdx.y = view
__global__ void degree_kernel(const int* __restrict__ ev,
                              float* __restrict__ deg, int M, int N) {
  int v = blockIdx.y;
  const int* dst = ev + (size_t)v * 2 * M + M;
  float* dv = deg + (size_t)v * N;
  long long stride = (long long)gridDim.x * blockDim.x;
  for (long long e = (long long)blockIdx.x * blockDim.x + threadIdx.x; e < M;
       e += stride)
    atomicAdd(&dv[dst[e]], 1.0f);
}

// nrm[v][e] = rsqrt(deg[src]) * rsqrt(deg[dst])
__global__ void norm_kernel(const int* __restrict__ ev,
                            const float* __restrict__ deg,
                            float* __restrict__ nrm, int M, int N) {
  int v = blockIdx.y;
  const int* src = ev + (size_t)v * 2 * M;
  const int* dst = src + M;
  const float* dv = deg + (size_t)v * N;
  float* nv = nrm + (size_t)v * M;
  long long stride = (long long)gridDim.x * blockDim.x;
  for (long long e = (long long)blockIdx.x * blockDim.x + threadIdx.x; e < M;
       e += stride)
    nv[e] = rsqrtf(dv[src[e]]) * rsqrtf(dv[dst[e]]);
}

// ---------------------------------------------------------------------------
// WMMA GEMM:  H[N x 64] = X[N x 128] @ W[128 x 64] + b
// Block = 128 threads = 4 waves; block owns one 16-row strip of X, staged
// once into LDS (async global->LDS when available), each wave computes one
// 16x16 output tile with exact-fp32 V_WMMA_F32_16X16X4_F32.
// LDS rows padded to 132 floats so a fragment read by lanes 0..15
// (row stride 132*4B) spreads across 16 distinct banks.
// ---------------------------------------------------------------------------
__global__ __launch_bounds__(128)
void gemm_wmma_kernel(const float* __restrict__ X, const float* __restrict__ W,
                      const float* __restrict__ B, float* __restrict__ H,
                      int Nrows) {
  __shared__ float ldsA[16][132];
  const int tid  = threadIdx.x;
  const int wave = tid >> 5;
  const int lane = tid & 31;
  const int row0 = blockIdx.x * 16;

  // ---- cooperative stage of the 16x128 A strip (512 float4 slots) ----
#pragma unroll
  for (int i = 0; i < 4; ++i) {
    int q  = tid + i * 128;      // 0..511
    int r  = q >> 5;             // row within strip
    int c4 = q & 31;             // float4 column
    int gr = row0 + r;
    if (gr >= Nrows) gr = Nrows - 1;   // clamp; EXEC stays full for WMMA
    const float* gsrc = X + (size_t)gr * IN_F + c4 * 4;
    float* ldst = &ldsA[r][c4 * 4];
#if defined(HAVE_ASYNC_LDS)
    __builtin_amdgcn_global_load_async_to_lds_b128(
        (as1_v4i*)(void*)const_cast<float*>(gsrc),
        (as3_v4i*)(void*)ldst, 0, 0);
#else
    *(float4*)ldst = *(const float4*)gsrc;
#endif
  }
#if defined(HAVE_ASYNC_LDS)
  __builtin_amdgcn_s_wait_asynccnt(0);
#endif
  __syncthreads();

  // ---- WMMA main loop ----
  const int col0 = wave * 16;
  const int m    = lane & 15;        // A row / B,D col within tile
  const int half = lane >> 4;
  const int koff = half << 1;        // lanes 16-31 hold K+2,K+3 (ISA layout)
  v8f c = {};
  for (int k = 0; k < IN_F; k += 4) {
    v2f a, bf;
    a.x  = ldsA[m][k + koff];                        // ds_load_b64
    a.y  = ldsA[m][k + koff + 1];
    bf.x = W[(size_t)(k + koff) * OUT_F + col0 + m]; // L2/L0-hot (32 KB)
    bf.y = W[(size_t)(k + koff + 1) * OUT_F + col0 + m];
    c = __builtin_amdgcn_wmma_f32_16x16x4_f32(false, a, false, bf, (short)0, c,
                                              false, false);
  }

  // C/D layout: VGPR r -> M=r (lanes 0-15), M=8+r (lanes 16-31); N = lane&15
  const float bias  = B[col0 + m];
  const int   rbase = half * 8;
#pragma unroll
  for (int r = 0; r < 8; ++r) {
    int row = row0 + rbase + r;
    if (row < Nrows) H[(size_t)row * OUT_F + col0 + m] = c[r] + bias;
  }
}

// ---------------------------------------------------------------------------
// agg[v][dst] += h[src] * norm   (16 float4 lanes per edge)
// ---------------------------------------------------------------------------
__global__ void agg_h_kernel(const int* __restrict__ ev,
                             const float* __restrict__ nrm,
                             const float* __restrict__ h,
                             float* __restrict__ agg, int M, int N) {
  int v = blockIdx.y;
  const int* src = ev + (size_t)v * 2 * M;
  const int* dst = src + M;
  const float* nv = nrm + (size_t)v * M;
  float* av = agg + (size_t)v * N * OUT_F;
  long long total = (long long)M * (OUT_F / 4);
  long long stride = (long long)gridDim.x * blockDim.x;
  for (long long idx = (long long)blockIdx.x * blockDim.x + threadIdx.x;
       idx < total; idx += stride) {
    int e  = (int)(idx >> 4);
    int f4 = (int)(idx & 15);
    float w = nv[e];
    const float4 hv = ((const float4*)(h + (size_t)src[e] * OUT_F))[f4];
    float* dp = av + (size_t)dst[e] * OUT_F + f4 * 4;
    atomicAdd(dp + 0, hv.x * w);
    atomicAdd(dp + 1, hv.y * w);
    atomicAdd(dp + 2, hv.z * w);
    atomicAdd(dp + 3, hv.w * w);
  }
}

// m_out[v*N + dst] += x[src] * norm   (32 float4 lanes per edge)
__global__ void agg_m_kernel(const int* __restrict__ ev,
                             const float* __restrict__ nrm,
                             const float* __restrict__ x,
                             float* __restrict__ m_out, int M, int N) {
  int v = blockIdx.y;
  const int* src = ev + (size_t)v * 2 * M;
  const int* dst = src + M;
  const float* nv = nrm + (size_t)v * M;
  float* mv = m_out + (size_t)v * N * IN_F;
  long long total = (long long)M * (IN_F / 4);
  long long stride = (long long)gridDim.x * blockDim.x;
  for (long long idx = (long long)blockIdx.x * blockDim.x + threadIdx.x;
       idx < total; idx += stride) {
    int e  = (int)(idx >> 5);
    int f4 = (int)(idx & 31);
    float w = nv[e];
    const float4 xv = ((const float4*)(x + (size_t)src[e] * IN_F))[f4];
    float* dp = mv + (size_t)dst[e] * IN_F + f4 * 4;
    atomicAdd(dp + 0, xv.x * w);
    atomicAdd(dp + 1, xv.y * w);
    atomicAdd(dp + 2, xv.z * w);
    atomicAdd(dp + 3, xv.w * w);
  }
}

// ---------------------------------------------------------------------------
// out_v = relu(agg_v + h * (1/deg_v)); write both the per-view block and its
// column slot of x_concat.
// ---------------------------------------------------------------------------
__global__ void finalize_x_kernel(const float* __restrict__ agg,
                                  const float* __restrict__ h,
                                  const float* __restrict__ deg,
                                  float* __restrict__ xcat,
                                  float* __restrict__ xview, int N) {
  long long total = (long long)N * OUT_F;
  long long stride = (long long)gridDim.x * blockDim.x;
  for (long long idx = (long long)blockIdx.x * blockDim.x + threadIdx.x;
       idx < total; idx += stride) {
    int i = (int)(idx >> 6);
    int j = (int)(idx & 63);
    float hij = h[idx];
#pragma unroll
    for (int v = 0; v < 4; ++v) {
      float invd = 1.0f / deg[(size_t)v * N + i];
      float val = agg[(size_t)v * N * OUT_F + idx] + hij * invd;
      val = fmaxf(val, 0.0f);
      xview[(size_t)v * N * OUT_F + idx] = val;
      xcat[(size_t)i * (4 * OUT_F) + v * OUT_F + j] = val;
    }
  }
}

__global__ void relu_kernel(float* __restrict__ p, long long n) {
  long long stride = (long long)gridDim.x * blockDim.x;
  for (long long i = (long long)blockIdx.x * blockDim.x + threadIdx.x; i < n;
       i += stride)
    p[i] = fmaxf(p[i], 0.0f);
}

// ---------------------------------------------------------------------------
static inline int nblk(long long total, int threads) {
  long long b = (total + threads - 1) / threads;
  if (b > 65535) b = 65535;
  if (b < 1) b = 1;
  return (int)b;
}

extern "C" void kernel_launch(void* const* d_in, const int* in_sizes, int n_in,
                              void* d_out, int out_size, void* d_ws,
                              size_t ws_size, hipStream_t stream) {
  const float* x  = (const float*)d_in[0];
  const float* W  = (const float*)d_in[1];
  const float* bb = (const float*)d_in[2];
  const int* pos  = (const int*)d_in[3];
  const int* neg  = (const int*)d_in[4];
  const int N = in_sizes[0] / IN_F;    // 100000
  const int M = in_sizes[3] / 2;       // 800000

  // workspace carve-up
  char* ws = (char*)d_ws;
  size_t off = 0;
  int* ev = (int*)(ws + off);          off += (size_t)4 * 2 * M * sizeof(int);
  float* deg = (float*)(ws + off);     off += (size_t)4 * N * sizeof(float);
  float* nrm = (float*)(ws + off);     off += (size_t)4 * M * sizeof(float);
  float* h   = (float*)(ws + off);     off += (size_t)N * OUT_F * sizeof(float);
  float* agg = (float*)(ws + off);     off += (size_t)4 * N * OUT_F * sizeof(float);
  (void)ws_size; (void)n_in; (void)out_size;

  // output carve-up (return order: m, x_concat, 4x per-view blocks)
  float* out   = (float*)d_out;
  float* m_out = out;                                  // [4N, 128]
  float* xcat  = out + (size_t)4 * N * IN_F;           // [N, 256]
  float* xview = xcat + (size_t)N * 4 * OUT_F;         // 4 x [N, 64]

  // 1) augmented edge sets
  build_edges_kernel<<<nblk(M, 256), 256, 0, stream>>>(pos, neg, ev, M, N);

  // 2) zero / init accumulators
  fill_kernel<<<nblk((long long)4 * N, 256), 256, 0, stream>>>(deg, 1.0f,
                                                               (long long)4 * N);
  fill_kernel<<<nblk((long long)4 * N * OUT_F, 256), 256, 0, stream>>>(
      agg, 0.0f, (long long)4 * N * OUT_F);
  fill_kernel<<<nblk((long long)4 * N * IN_F, 256), 256, 0, stream>>>(
      m_out, 0.0f, (long long)4 * N * IN_F);

  // 3) degrees + per-edge norms
  {
    dim3 g(nblk(M, 256), 4);
    degree_kernel<<<g, 256, 0, stream>>>(ev, deg, M, N);
    norm_kernel<<<g, 256, 0, stream>>>(ev, deg, nrm, M, N);
  }

  // 4) h = x @ W + b via fp32 WMMA (computed ONCE; identical for all 8
  //    reference gcn_conv calls)
  gemm_wmma_kernel<<<(N + 15) / 16, 128, 0, stream>>>(x, W, bb, h, N);

  // 5) edge-wise gather/scatter aggregation (L2-resident x/h), float4 gathers
  {
    dim3 gh(nblk((long long)M * (OUT_F / 4), 256), 4);
    agg_h_kernel<<<gh, 256, 0, stream>>>(ev, nrm, h, agg, M, N);
    dim3 gm(nblk((long long)M * (IN_F / 4), 256), 4);
    agg_m_kernel<<<gm, 256, 0, stream>>>(ev, nrm, x, m_out, M, N);
  }

  // 6) finalize: self-loop term + relu; build x_concat and per-view outputs
  finalize_x_kernel<<<nblk((long long)N * OUT_F, 256), 256, 0, stream>>>(
      agg, h, deg, xcat, xview, N);
  relu_kernel<<<nblk((long long)4 * N * IN_F, 256), 256, 0, stream>>>(
      m_out, (long long)4 * N * IN_F);
}